// SimpleCVRPSolver_86887188398503
// MI455X (gfx1250) — compile-verified
//
#include <hip/hip_runtime.h>
#include <hip/hip_bf16.h>
#include <math.h>

typedef float v2f __attribute__((ext_vector_type(2)));
typedef float v8f __attribute__((ext_vector_type(8)));

#define NEGV   (-1e30f)
#define TINYF  (1.1754943508222875e-38f)

// ---------------------------------------------------------------------------
// fp32 WMMA 16x16 tile GEMM helper (CDNA5 V_WMMA_F32_16X16X4_F32).
// A: row-major [16 x K] at stride lda (pre-offset to tile row 0).
// B: element B[k][n] = Bptr[k*ldbk + n*ldbn] (pre-offset to tile col 0).
// Layouts (wave32): A lane r=M, half selects K pair {k0+2h, k0+2h+1};
// B lane r=N, same K-pair rule; C VGPR i -> row i+8*half, col r.
// ---------------------------------------------------------------------------
__device__ __forceinline__ v8f wmma_tile_f32(const float* A, int lda,
                                             const float* B, int ldbk, int ldbn,
                                             int K, unsigned lane) {
  const unsigned r  = lane & 15u;
  const unsigned h2 = (lane >> 4) & 1u;
  v8f c = {};
  const float* arow = A + (int)r * lda;
  const float* bcol = B + (int)r * ldbn;
  for (int k0 = 0; k0 < K; k0 += 4) {
    const int ka = k0 + 2 * (int)h2;
    v2f a, b;
    a.x = arow[ka];
    a.y = arow[ka + 1];
    b.x = bcol[ka * ldbk];
    b.y = bcol[(ka + 1) * ldbk];
    c = __builtin_amdgcn_wmma_f32_16x16x4_f32(false, a, false, b, (short)0, c,
                                              false, false);
  }
  return c;
}

// ---------------------------------------------------------------------------
// Kernel 1: mask-independent pointer-network scores s[B][N], one block/batch.
// B=64, N=100 (padded 112), H=64, heads=4, dh=16. 128 threads = 4 waves.
// ---------------------------------------------------------------------------
__global__ __launch_bounds__(128)
void pointer_scores_kernel(const float* __restrict__ nf,     // [64,100,3]
                           const float* __restrict__ W_emb,  // [64,3]
                           const float* __restrict__ b_emb,  // [64]
                           const float* __restrict__ Wio,    // [192,64]
                           const float* __restrict__ bio,    // [192]
                           const float* __restrict__ Wo,     // [64,64]
                           const float* __restrict__ bo,     // [64]
                           const float* __restrict__ W1,     // [64,128]
                           const float* __restrict__ b1,     // [64]
                           const float* __restrict__ W2,     // [1,64]
                           const float* __restrict__ b2,     // [1]
                           float* __restrict__ scores)       // [64,100]
{
  constexpr int N = 100, NP = 112, H = 64;
  const int b   = blockIdx.x;
  const int tid = threadIdx.x;
  const unsigned lane = (unsigned)tid & 31u;
  const int wave = tid >> 5;                 // 0..3
  const unsigned r  = lane & 15u;
  const unsigned h2 = (lane >> 4) & 1u;

  extern __shared__ float smem[];
  float* sEmb = smem;                        // [112][68]
  float* sQ   = sEmb + NP * 68;              // [112][20]
  float* sK   = sQ   + NP * 20;              // [112][20]
  float* sV   = sK   + NP * 20;              // [112][20]
  float* sAtt = sV   + NP * 20;              // [112][68]
  float* sWS  = sAtt + NP * 68;              // [4 waves][16][116]
  float* sCtx = sWS  + 4 * 16 * 116;         // [64]
  float* sC2  = sCtx + 64;                   // [64]
  float* sOut = sEmb;                        // alias: emb dead after head loop
  float* sH   = sAtt;                        // alias: attended dead after out

  // zero emb buffer (padding rows 100..111 must be 0)
  for (int i = tid; i < NP * 68; i += 128) sEmb[i] = 0.0f;
  __syncthreads();
  // emb = nf @ W_emb^T + b_emb  (K=3, plain VALU)
  for (int i = tid; i < N * H; i += 128) {
    int n = i >> 6, k = i & 63;
    const float* x = nf + (b * N + n) * 3;
    sEmb[n * 68 + k] = x[0] * W_emb[k * 3 + 0] + x[1] * W_emb[k * 3 + 1] +
                       x[2] * W_emb[k * 3 + 2] + b_emb[k];
  }
  __syncthreads();

  for (int hd = 0; hd < 4; ++hd) {
    // q,k,v [112][16] for this head: 21 jobs = part(3) x mtile(7), K=64
    for (int job = wave; job < 21; job += 4) {
      int part = job / 7, mt = job % 7;
      int wro = part * 64 + hd * 16;                    // in_proj row offset
      v8f c = wmma_tile_f32(sEmb + mt * 16 * 68, 68, Wio + wro * 64, 1, 64,
                            64, lane);
      float bias = bio[wro + (int)r];
      float* dst = (part == 0) ? sQ : (part == 1) ? sK : sV;
      for (int i = 0; i < 8; ++i)
        dst[(mt * 16 + i + 8 * (int)h2) * 20 + (int)r] = c[i] + bias;
    }
    __syncthreads();
    // attention, one m-tile of 16 query rows per wave (wave-private LDS)
    for (int mt = wave; mt < 7; mt += 4) {
      float* ws = sWS + wave * 16 * 116;
      for (int nt = 0; nt < 7; ++nt) {
        v8f c = wmma_tile_f32(sQ + mt * 16 * 20, 20, sK + nt * 16 * 20, 1, 20,
                              16, lane);
        for (int i = 0; i < 8; ++i)
          ws[(i + 8 * (int)h2) * 116 + nt * 16 + (int)r] = c[i];
      }
      asm volatile("" ::: "memory");  // LDS in-order per wave; stop reordering
      if (lane < 16u) {               // row-wise softmax over n<100, scale 1/4
        float* row = ws + (int)lane * 116;
        float mx = -3.4e38f;
        for (int n = 0; n < N; ++n) mx = fmaxf(mx, row[n] * 0.25f);
        float s = 0.0f;
        for (int n = 0; n < N; ++n) {
          float e = expf(row[n] * 0.25f - mx);
          row[n] = e; s += e;
        }
        float inv = 1.0f / s;
        for (int n = 0; n < N; ++n) row[n] *= inv;
        for (int n = N; n < NP; ++n) row[n] = 0.0f;
      }
      asm volatile("" ::: "memory");
      // attended[16][16] = attn[16][112] @ v[112][16]  (K=112)
      v8f c = wmma_tile_f32(ws, 116, sV, 20, 1, NP, lane);
      for (int i = 0; i < 8; ++i)
        sAtt[(mt * 16 + i + 8 * (int)h2) * 68 + hd * 16 + (int)r] = c[i];
    }
    __syncthreads();
  }

  // out = attended @ out_proj^T + bo : 28 jobs = mtile(7) x ntile(4), K=64
  for (int job = wave; job < 28; job += 4) {
    int mt = job >> 2, nt = job & 3;
    v8f c = wmma_tile_f32(sAtt + mt * 16 * 68, 68, Wo + nt * 16 * 64, 1, 64,
                          64, lane);
    float bias = bo[nt * 16 + (int)r];
    for (int i = 0; i < 8; ++i)
      sOut[(mt * 16 + i + 8 * (int)h2) * 68 + nt * 16 + (int)r] = c[i] + bias;
  }
  __syncthreads();
  // context = mean over n<100 of out; c2 = b1 + context @ W1[:,64:]^T
  if (tid < 64) {
    float s = 0.0f;
    for (int n = 0; n < N; ++n) s += sOut[n * 68 + tid];
    sCtx[tid] = s * (1.0f / (float)N);
  }
  __syncthreads();
  if (tid < 64) {
    float s = b1[tid];
    for (int k = 0; k < 64; ++k) s += sCtx[k] * W1[tid * 128 + 64 + k];
    sC2[tid] = s;
  }
  __syncthreads();
  // h = relu(out @ W1[:, :64]^T + c2)
  for (int job = wave; job < 28; job += 4) {
    int mt = job >> 2, nt = job & 3;
    v8f c = wmma_tile_f32(sOut + mt * 16 * 68, 68, W1 + nt * 16 * 128, 1, 128,
                          64, lane);
    for (int i = 0; i < 8; ++i) {
      float v = c[i] + sC2[nt * 16 + (int)r];
      sH[(mt * 16 + i + 8 * (int)h2) * 68 + nt * 16 + (int)r] = fmaxf(v, 0.0f);
    }
  }
  __syncthreads();
  // s[m] = h[m] . W2 + b2
  for (int m = tid; m < N; m += 128) {
    float acc = b2[0];
    for (int j = 0; j < 64; ++j) acc += sH[m * 68 + j] * W2[j];
    scores[b * N + m] = acc;
  }
}

// ---------------------------------------------------------------------------
// JAX threefry2x32 (Random123), exact rotation/key schedule.
// ---------------------------------------------------------------------------
__device__ __forceinline__ void threefry2x32(unsigned k0, unsigned k1,
                                             unsigned x0, unsigned x1,
                                             unsigned& o0, unsigned& o1) {
  const unsigned ks2 = k0 ^ k1 ^ 0x1BD11BDAu;
  const unsigned ks[3] = {k0, k1, ks2};
  const unsigned rotsA[4] = {13u, 15u, 26u, 6u};
  const unsigned rotsB[4] = {17u, 29u, 16u, 24u};
  unsigned a = x0 + k0, bb = x1 + k1;
#pragma unroll
  for (unsigned i = 0; i < 5; ++i) {
    const unsigned* rr = (i & 1u) ? rotsB : rotsA;
#pragma unroll
    for (int j = 0; j < 4; ++j) {
      a += bb;
      bb = (bb << rr[j]) | (bb >> (32u - rr[j]));
      bb ^= a;
    }
    a  += ks[(i + 1u) % 3u];
    bb += ks[(i + 2u) % 3u] + (i + 1u);
  }
  o0 = a; o1 = bb;
}

// ---------------------------------------------------------------------------
// Kernel 2: 40-step sequential decode. One block of 1024 threads:
// 16 lanes per batch (64 batches), shuffle reductions within each 16-group,
// LDS atomicAnd for the global all(visited) flag.
// ---------------------------------------------------------------------------
__global__ __launch_bounds__(1024)
void decode_kernel(const float* __restrict__ scores,   // [64,100] from d_ws
                   const float* __restrict__ demands,  // [64,100]
                   const float* __restrict__ caps,     // [64]
                   const int* __restrict__ msPtr,      // scalar max_steps
                   void* __restrict__ outBase)         // routes i32 + logp f32
{
  __shared__ int sFlag;
  const int tid = threadIdx.x;
  const int b   = tid >> 4;
  const int sub = tid & 15;
  const int ms  = msPtr[0];
  int*   routes = (int*)outBase;
  float* lsum   = (float*)outBase + 64 * ms;

  const float cap = caps[b];
  float sv[7], dv[7];
#pragma unroll
  for (int i = 0; i < 7; ++i) {
    int n = sub + 16 * i;
    if (n < 100) { sv[i] = scores[b * 100 + n]; dv[i] = demands[b * 100 + n]; }
    else         { sv[i] = 0.0f; dv[i] = 0.0f; }
  }
  unsigned vis = (sub == 0) ? 1u : 0u;   // bit i <-> node sub+16*i; depot set
  float rem = cap;
  float lp  = 0.0f;
  bool active = true;

  for (int step = 0; step < ms; ++step) {
    // key_t = fold_in(key(42), t) = threefry([0,42],[0,t])
    unsigned kk0, kk1;
    threefry2x32(0u, 42u, 0u, (unsigned)step, kk0, kk1);

    float lg[7];
    float mx = -3.4e38f;
#pragma unroll
    for (int i = 0; i < 7; ++i) {
      int n = sub + 16 * i;
      if (n < 100) {
        bool msk = (((vis >> i) & 1u) || (dv[i] > rem)) && (n != 0);
        lg[i] = msk ? NEGV : sv[i];
        mx = fmaxf(mx, lg[i]);
      } else lg[i] = -3.4e38f;
    }
    for (int off = 1; off < 16; off <<= 1)
      mx = fmaxf(mx, __shfl_xor(mx, off, 32));
    float se = 0.0f;
#pragma unroll
    for (int i = 0; i < 7; ++i)
      if (sub + 16 * i < 100) se += expf(lg[i] - mx);
    for (int off = 1; off < 16; off <<= 1)
      se += __shfl_xor(se, off, 32);
    const float lse = logf(se);

    // gumbel noise (JAX random_bits split-counter layout) + first-max argmax
    float best = -3.4e38f; int bidx = 1 << 30; float blp = 0.0f;
#pragma unroll
    for (int i = 0; i < 7; ++i) {
      int n = sub + 16 * i;
      if (n >= 100) continue;
      int idx = b * 100 + n;                 // flat index into [64,100]
      unsigned o0, o1, bits;
      if (idx < 3200) { threefry2x32(kk0, kk1, (unsigned)idx, (unsigned)(idx + 3200), o0, o1); bits = o0; }
      else            { threefry2x32(kk0, kk1, (unsigned)(idx - 3200), (unsigned)idx, o0, o1); bits = o1; }
      float f = __uint_as_float((bits >> 9) | 0x3f800000u) - 1.0f;
      float u = fmaxf(TINYF, f * (1.0f - TINYF) + TINYF);
      float g = -logf(-logf(u));
      float l = lg[i] - mx - lse;            // log_prob
      float z = l + g;
      if (z > best) { best = z; bidx = n; blp = l; }
    }
    for (int off = 1; off < 16; off <<= 1) {
      float ob = __shfl_xor(best, off, 32);
      int   oi = __shfl_xor(bidx, off, 32);
      float ol = __shfl_xor(blp,  off, 32);
      if (ob > best || (ob == best && oi < bidx)) { best = ob; bidx = oi; blp = ol; }
    }
    const int nxt = bidx;

    // chosen node's demand via masked sum-reduce (avoids dynamic indexing)
    float dsel = 0.0f;
#pragma unroll
    for (int i = 0; i < 7; ++i) {
      int n = sub + 16 * i;
      if (n < 100 && n == nxt) dsel = dv[i];
    }
    for (int off = 1; off < 16; off <<= 1)
      dsel += __shfl_xor(dsel, off, 32);

    if (sub == 0) routes[b * ms + step] = active ? nxt : -1;
    lp += active ? blp : 0.0f;

    const bool isDepot = (nxt == 0);
    if (isDepot) vis = (sub == 0) ? 1u : 0u;
    else {
#pragma unroll
      for (int i = 0; i < 7; ++i)
        if (sub + 16 * i == nxt) vis |= (1u << i);
    }
    rem = isDepot ? cap : (rem - dsel);

    // global all(visited[:,1:]) across every batch
    int ok = 1;
#pragma unroll
    for (int i = 0; i < 7; ++i) {
      int n = sub + 16 * i;
      if (n >= 1 && n < 100 && !((vis >> i) & 1u)) ok = 0;
    }
    for (int off = 1; off < 16; off <<= 1)
      ok &= __shfl_xor(ok, off, 32);
    __syncthreads();
    if (tid == 0) sFlag = 1;
    __syncthreads();
    if (sub == 0) atomicAnd(&sFlag, ok);
    __syncthreads();
    active = active && (sFlag == 0);
  }
  if (sub == 0) lsum[b] = lp;
}

// ---------------------------------------------------------------------------
extern "C" void kernel_launch(void* const* d_in, const int* in_sizes, int n_in,
                              void* d_out, int out_size, void* d_ws, size_t ws_size,
                              hipStream_t stream) {
  const float* nf    = (const float*)d_in[0];
  const float* dem   = (const float*)d_in[1];
  const float* caps  = (const float*)d_in[2];
  const float* W_emb = (const float*)d_in[3];
  const float* b_emb = (const float*)d_in[4];
  const float* Wio   = (const float*)d_in[5];
  const float* bio   = (const float*)d_in[6];
  const float* Wo    = (const float*)d_in[7];
  const float* bo    = (const float*)d_in[8];
  const float* W1    = (const float*)d_in[9];
  const float* b1    = (const float*)d_in[10];
  const float* W2    = (const float*)d_in[11];
  const float* b2    = (const float*)d_in[12];
  const int*   msp   = (const int*)d_in[13];

  float* scores = (float*)d_ws;   // 64*100 fp32 scratch

  const size_t shbytes =
      (size_t)(112 * 68 + 3 * 112 * 20 + 112 * 68 + 4 * 16 * 116 + 128) *
      sizeof(float);              // ~118 KB dynamic LDS (<< 320 KB/WGP)

  pointer_scores_kernel<<<64, 128, shbytes, stream>>>(
      nf, W_emb, b_emb, Wio, bio, Wo, bo, W1, b1, W2, b2, scores);
  decode_kernel<<<1, 1024, 0, stream>>>(scores, dem, caps, msp, d_out);
}